// ExpertPreferredRouter_70746701300041
// MI455X (gfx1250) — compile-verified
//
#include <hip/hip_runtime.h>
#include <hip/hip_bf16.h>

typedef __attribute__((ext_vector_type(2))) float v2f;
typedef __attribute__((ext_vector_type(4))) float v4f;
typedef __attribute__((ext_vector_type(8))) float v8f;

#define D_DIM 1024
#define E_NUM 4
#define N_TOK 8192
#define B_NUM 8

// ---------------------------------------------------------------------------
// Kernel 1: router logits via V_WMMA_F32_16X16X4_F32 + fused softmax.
// 8 waves / block, each wave owns a 16-token tile.
// K remap: step t uses columns {2t, 2t+1, 512+2t, 512+2t+1} so lanes 0-15
// stream D-half [0,512) and lanes 16-31 stream [512,1024) contiguously:
// one b128 load feeds two WMMAs. B pad columns 4..15 read duplicated expert
// rows (col & 3) -- their C columns are never consumed, so no zeroing and no
// EXEC divergence in the loop. Two accumulators break the XDL RAW chain.
// ---------------------------------------------------------------------------
__global__ __launch_bounds__(256) void router_logits_softmax(
    const float* __restrict__ x, const float* __restrict__ W,
    const float* __restrict__ bias, float* __restrict__ probs)
{
    __shared__ float sW[E_NUM * D_DIM];  // 16 KB
    const int tid = threadIdx.x;
    for (int i = tid; i < E_NUM * D_DIM; i += 256) sW[i] = W[i];
    __syncthreads();

    const int lane = tid & 31;
    const int wave = tid >> 5;
    const size_t tok0 = ((size_t)blockIdx.x * 8 + wave) * 16;

    const int row  = lane & 15;        // A: M index
    const int col  = lane & 15;        // C/D: N index (expert, valid < 4)
    const int colw = lane & 3;         // duplicated W row for pad columns
    const int half = lane >> 4;        // K-half select / C M-offset

    const float* xrow = x + (tok0 + row) * D_DIM + half * (D_DIM / 2);
    const float* wrow = &sW[colw * D_DIM + half * (D_DIM / 2)];

    v8f acc0 = {};
    v8f acc1 = {};
    #pragma unroll 4
    for (int u = 0; u < D_DIM / 2; u += 4) {
        v4f a4 = *(const v4f*)(xrow + u);   // global_load_b128: 2 steps of A
        v4f b4 = *(const v4f*)(wrow + u);   // ds_load_b128:     2 steps of B
        v2f a0; a0.x = a4.x; a0.y = a4.y;
        v2f a1; a1.x = a4.z; a1.y = a4.w;
        v2f b0; b0.x = b4.x; b0.y = b4.y;
        v2f b1; b1.x = b4.z; b1.y = b4.w;
        acc0 = __builtin_amdgcn_wmma_f32_16x16x4_f32(
            false, a0, false, b0, (short)0, acc0, false, false);
        acc1 = __builtin_amdgcn_wmma_f32_16x16x4_f32(
            false, a1, false, b1, (short)0, acc1, false, false);
    }

    const float bv = bias[lane & 3];
    #pragma unroll
    for (int r = 0; r < 8; ++r) {
        // token (tok0 + r + 8*half): experts 0..3 live in aligned lane group of 4
        float v = (acc0[r] + acc1[r]) + bv;
        float m = v;
        m = fmaxf(m, __shfl_xor(m, 1));
        m = fmaxf(m, __shfl_xor(m, 2));
        float e = __expf(v - m);
        float s = e;
        s += __shfl_xor(s, 1);
        s += __shfl_xor(s, 2);
        float p = e / s;
        if (col < E_NUM) {
            size_t tok = tok0 + (size_t)r + (size_t)half * 8;
            probs[tok * E_NUM + col] = p;
        }
    }
}

// ---------------------------------------------------------------------------
// Kernel 2: capacity-constrained expert-preferred assignment.
// One 1024-thread block per batch row. For j = 3..0: build 64-bit keys
// (probBits<<13 | (8191-idx)) for unassigned tokens, bitonic-sort descending
// in LDS (8192 x u64 = 64 KB), take top k[j]. Leftover token -> expert 0.
// ---------------------------------------------------------------------------
__global__ __launch_bounds__(1024) void route_assign(
    const float* __restrict__ probs, unsigned char* __restrict__ flags,
    float* __restrict__ out_mask, float* __restrict__ out_prob)
{
    extern __shared__ unsigned long long keys[];  // N_TOK entries (dynamic 64 KB)
    const int b   = blockIdx.x;
    const int tid = threadIdx.x;
    const float* P = probs + (size_t)b * N_TOK * E_NUM;
    unsigned char* F = flags + (size_t)b * N_TOK;
    float* Mo = out_mask + (size_t)b * N_TOK;
    float* Qo = out_prob + (size_t)b * N_TOK;

    for (int i = tid; i < N_TOK; i += 1024) F[i] = 0;
    __syncthreads();

    const int kcap[E_NUM] = {819, 1228, 2048, 4096};  // floor(CAP * N)
    for (int j = E_NUM - 1; j >= 0; --j) {
        // positive-float bit pattern preserves order; low 13 bits encode
        // (8191 - idx) so ties resolve to the lower token index first.
        for (int i = tid; i < N_TOK; i += 1024) {
            unsigned int pb = __float_as_uint(P[(size_t)i * E_NUM + j]);
            unsigned long long key =
                (((unsigned long long)pb) << 13) |
                (unsigned long long)(N_TOK - 1 - i);
            keys[i] = F[i] ? 0ull : key;
        }
        __syncthreads();

        // bitonic sort, descending
        for (int ksz = 2; ksz <= N_TOK; ksz <<= 1) {
            for (int s = ksz >> 1; s > 0; s >>= 1) {
                for (int i = tid; i < N_TOK; i += 1024) {
                    int ixj = i ^ s;
                    if (ixj > i) {
                        unsigned long long a = keys[i];
                        unsigned long long c = keys[ixj];
                        bool desc = ((i & ksz) == 0);
                        if (desc ? (a < c) : (a > c)) {
                            keys[i]   = c;
                            keys[ixj] = a;
                        }
                    }
                }
                __syncthreads();
            }
        }

        // claim top k[j] tokens for expert j
        for (int i = tid; i < kcap[j]; i += 1024) {
            int tok = N_TOK - 1 - (int)(keys[i] & 0x1FFFull);
            F[tok]  = 1;
            Mo[tok] = (float)j;
            Qo[tok] = P[(size_t)tok * E_NUM + j];
        }
        __syncthreads();
    }

    // single leftover token (sum(k)=8191) -> expert 0
    for (int i = tid; i < N_TOK; i += 1024) {
        if (!F[i]) {
            Mo[i] = 0.0f;
            Qo[i] = P[(size_t)i * E_NUM + 0];
        }
    }
}

extern "C" void kernel_launch(void* const* d_in, const int* in_sizes, int n_in,
                              void* d_out, int out_size, void* d_ws, size_t ws_size,
                              hipStream_t stream) {
    (void)in_sizes; (void)n_in; (void)out_size; (void)ws_size;
    const float* x    = (const float*)d_in[0];  // [B, N, D] f32
    const float* W    = (const float*)d_in[1];  // [E, D]   f32
    const float* bias = (const float*)d_in[2];  // [E]      f32

    float* out_mask = (float*)d_out;                          // [B*N] token_mask
    float* out_prob = (float*)d_out + (size_t)B_NUM * N_TOK;  // [B*N] expert_probs

    float* probs = (float*)d_ws;                              // [B*N, E] f32 (1 MB)
    unsigned char* flags =
        (unsigned char*)d_ws + (size_t)B_NUM * N_TOK * E_NUM * sizeof(float);

    const int total_tokens = B_NUM * N_TOK;                   // 65536
    const int blocks = total_tokens / 128;                    // 512 (128 tokens/block)
    router_logits_softmax<<<blocks, 256, 0, stream>>>(x, W, bias, probs);

    route_assign<<<B_NUM, 1024, N_TOK * sizeof(unsigned long long), stream>>>(
        probs, flags, out_mask, out_prob);
}